// MinLSTMCell_log_26345329394157
// MI455X (gfx1250) — compile-verified
//
#include <hip/hip_runtime.h>
#include <hip/hip_bf16.h>

// ---------------------------------------------------------------------------
// minLSTM (log-space parallel scan) for MI455X / gfx1250, wave32, WMMA f32.
//
//   K1 gates_gemm : f/i/h pre-activations via V_WMMA_F32_16X16X4_F32,
//                   LDS-staged double-buffered 64x64 block tiles,
//                   fused epilogue -> log_f, log_v = log_i + log_g(h_pre)
//   K2 scan_local : chunk-local logaddexp scan (chunk = 256)
//   K3 scan_chunk : scan of 16 chunk summaries per (b,h); writes h_last
//   K4 scan_apply : apply carries, h = exp(...)  (in place into log_f buffer)
//   K5 out_gemm   : out = h @ Wo + bo, same LDS-staged WMMA structure
// ---------------------------------------------------------------------------

typedef __attribute__((ext_vector_type(2))) float v2f;
typedef __attribute__((ext_vector_type(4))) float v4f;
typedef __attribute__((ext_vector_type(8))) float v8f;

#define BB   8
#define SS   4096
#define EE   512
#define HH   512
#define MM   (BB * SS)        // 32768 rows
#define CS   256              // scan chunk length
#define NC   (SS / CS)        // 16 chunks
#define BSH  ((size_t)MM * HH)
#define BNH  ((size_t)BB * NC * HH)

#define KC    16              // GEMM K-chunk staged in LDS
#define NKC   (EE / KC)       // 32 chunks
#define APAD  20              // padded LDS stride for A tiles (bank-friendly)

#define NEG_BIG (-1.0e30f)

__device__ __forceinline__ float softplusf(float x) {
    return fmaxf(x, 0.0f) + log1pf(expf(-fabsf(x)));
}
__device__ __forceinline__ float log_gf(float x) {
    return (x >= 0.0f) ? logf(x + 0.5f) : -softplusf(-x);
}
__device__ __forceinline__ float logaddexpf_(float a, float b) {
    float m = fmaxf(a, b);
    if (m <= NEG_BIG) return NEG_BIG;
    return m + log1pf(expf(fminf(a, b) - m));
}

__device__ __forceinline__ v8f wmma_f32(v2f a, v2f b, v8f c) {
    return __builtin_amdgcn_wmma_f32_16x16x4_f32(false, a, false, b, (short)0, c,
                                                 false, false);
}

// ---------------------------------------------------------------------------
// K1: gate GEMMs + fused log-space epilogue.
// 256 threads = 8 waves (2 M x 4 N). Block tile 64(M) x 64(N).
// LDS double-buffered: A[64x16], B[3][16x64] per buffer.
// Each wave: two 16x16 M-subtiles x 3 gates = 6 v8f accumulators.
// ---------------------------------------------------------------------------
__global__ __launch_bounds__(256) void gates_gemm_kernel(
    const float* __restrict__ X,
    const float* __restrict__ Wf, const float* __restrict__ bf,
    const float* __restrict__ Wi, const float* __restrict__ bi,
    const float* __restrict__ Wh, const float* __restrict__ bh,
    float* __restrict__ logf_buf, float* __restrict__ logv_buf)
{
    __shared__ float As[2][64][APAD];
    __shared__ float Bs[2][3][KC][64];

    const int lane = threadIdx.x & 31;
    const int wv   = threadIdx.x >> 5;
    const int wm   = wv >> 2;            // 0..1  (M sub-block of 32)
    const int wn   = wv & 3;             // 0..3  (N sub-block of 16)
    const int lm   = lane & 15;
    const int khalf = (lane >> 4) << 1;  // 0 or 2 (A/B k-lane split)

    const int mblock = blockIdx.x * 64;
    const int col0   = blockIdx.y * 64;
    const int cn     = wn * 16 + lm;     // column within block tile
    const int col    = col0 + cn;        // global N index

    // staging thread mapping
    const int ar = threadIdx.x >> 2;           // A row 0..63
    const int ak = (threadIdx.x & 3) << 2;     // A k   0,4,8,12
    const int bk = threadIdx.x >> 4;           // B k-row 0..15
    const int bc = (threadIdx.x & 15) << 2;    // B col 0,4,...,60

    v8f aF0 = {}, aI0 = {}, aH0 = {};
    v8f aF1 = {}, aI1 = {}, aH1 = {};

    // prologue stage chunk 0
    {
        const int k0 = 0;
        *(v4f*)&As[0][ar][ak] = *(const v4f*)(X + (size_t)(mblock + ar) * EE + k0 + ak);
        const size_t wofs = (size_t)(k0 + bk) * HH + col0 + bc;
        *(v4f*)&Bs[0][0][bk][bc] = *(const v4f*)(Wf + wofs);
        *(v4f*)&Bs[0][1][bk][bc] = *(const v4f*)(Wi + wofs);
        *(v4f*)&Bs[0][2][bk][bc] = *(const v4f*)(Wh + wofs);
    }
    __syncthreads();

    for (int kc = 0; kc < NKC; ++kc) {
        const int cur = kc & 1;
        if (kc + 1 < NKC) {
            const int k0 = (kc + 1) * KC;
            __builtin_prefetch(X + (size_t)(mblock + ar) * EE + k0 + KC, 0, 0);
            *(v4f*)&As[cur ^ 1][ar][ak] =
                *(const v4f*)(X + (size_t)(mblock + ar) * EE + k0 + ak);
            const size_t wofs = (size_t)(k0 + bk) * HH + col0 + bc;
            *(v4f*)&Bs[cur ^ 1][0][bk][bc] = *(const v4f*)(Wf + wofs);
            *(v4f*)&Bs[cur ^ 1][1][bk][bc] = *(const v4f*)(Wi + wofs);
            *(v4f*)&Bs[cur ^ 1][2][bk][bc] = *(const v4f*)(Wh + wofs);
        }
#pragma unroll
        for (int kk = 0; kk < KC; kk += 4) {
            const int kh = kk + khalf;
            v2f a0 = *(const v2f*)&As[cur][wm * 32 + lm][kh];
            v2f a1 = *(const v2f*)&As[cur][wm * 32 + 16 + lm][kh];
            v2f b0, b1, b2;
            b0.x = Bs[cur][0][kh][cn];     b0.y = Bs[cur][0][kh + 1][cn];
            b1.x = Bs[cur][1][kh][cn];     b1.y = Bs[cur][1][kh + 1][cn];
            b2.x = Bs[cur][2][kh][cn];     b2.y = Bs[cur][2][kh + 1][cn];
            aF0 = wmma_f32(a0, b0, aF0);
            aI0 = wmma_f32(a0, b1, aI0);
            aH0 = wmma_f32(a0, b2, aH0);
            aF1 = wmma_f32(a1, b0, aF1);
            aI1 = wmma_f32(a1, b1, aI1);
            aH1 = wmma_f32(a1, b2, aH1);
        }
        __syncthreads();
    }

    // epilogue: bias + gate math -> log_f, log_v
    const float bfv = bf[col], biv = bi[col], bhv = bh[col];
#pragma unroll
    for (int sub = 0; sub < 2; ++sub) {
        const v8f accF = sub ? aF1 : aF0;
        const v8f accI = sub ? aI1 : aI0;
        const v8f accH = sub ? aH1 : aH0;
        const int mbase = mblock + wm * 32 + sub * 16 + ((lane >> 4) << 3);
#pragma unroll
        for (int r = 0; r < 8; ++r) {
            const int mrow = mbase + r;
            const float f  = accF[r] + bfv;
            const float i  = accI[r] + biv;
            const float hp = accH[r] + bhv;
            const float diff = softplusf(-f) - softplusf(-i);
            const float lf = -softplusf(diff);
            const float li = -softplusf(-diff);
            const float lv = li + log_gf(hp);
            const size_t idx = (size_t)mrow * HH + col;
            logf_buf[idx] = lf;
            logv_buf[idx] = lv;
        }
    }
}

// ---------------------------------------------------------------------------
// K2: chunk-local scan. One thread per (b, chunk, h). Coalesced over h.
// ---------------------------------------------------------------------------
__global__ __launch_bounds__(256) void scan_local_kernel(
    const float* __restrict__ logf_buf, float* __restrict__ logv_buf,
    float* __restrict__ sumA, float* __restrict__ sumL)
{
    const int tid = blockIdx.x * blockDim.x + threadIdx.x;  // (b*NC + c)*HH + h
    const int h  = tid % HH;
    const int bc = tid / HH;
    const int b  = bc / NC;
    const int c  = bc % NC;
    const size_t base = ((size_t)(b * SS + c * CS)) * HH + h;

    float A = 0.0f, L = NEG_BIG;
    for (int t = 0; t < CS; ++t) {
        const size_t idx = base + (size_t)t * HH;
        const float lf = logf_buf[idx];
        const float lv = logv_buf[idx];
        A += lf;
        L = logaddexpf_(L + lf, lv);
        logv_buf[idx] = L;
    }
    sumA[(size_t)bc * HH + h] = A;
    sumL[(size_t)bc * HH + h] = L;
}

// ---------------------------------------------------------------------------
// K3: scan across chunk summaries; produce per-chunk carry-in and h_last.
// ---------------------------------------------------------------------------
__global__ __launch_bounds__(256) void scan_chunks_kernel(
    const float* __restrict__ h0,
    const float* __restrict__ sumA, const float* __restrict__ sumL,
    float* __restrict__ carryin, float* __restrict__ hlast)
{
    const int tid = blockIdx.x * blockDim.x + threadIdx.x;  // 0 .. BB*HH-1
    const int b = tid / HH;
    const int h = tid % HH;
    float carry = log_gf(h0[(size_t)b * HH + h]);
    for (int c = 0; c < NC; ++c) {
        const size_t i = ((size_t)(b * NC + c)) * HH + h;
        carryin[i] = carry;
        carry = logaddexpf_(carry + sumA[i], sumL[i]);
    }
    hlast[(size_t)b * HH + h] = expf(carry);
}

// ---------------------------------------------------------------------------
// K4: apply carries: h_t = exp(logaddexp(prefix_a_t + carry, L_local_t)).
// Writes h in place into logf_buf.
// ---------------------------------------------------------------------------
__global__ __launch_bounds__(256) void scan_apply_kernel(
    float* __restrict__ logf_buf, const float* __restrict__ logv_buf,
    const float* __restrict__ carryin)
{
    const int tid = blockIdx.x * blockDim.x + threadIdx.x;
    const int h  = tid % HH;
    const int bc = tid / HH;
    const int b  = bc / NC;
    const int c  = bc % NC;
    const size_t base = ((size_t)(b * SS + c * CS)) * HH + h;

    const float carry = carryin[(size_t)bc * HH + h];
    float a = 0.0f;
    for (int t = 0; t < CS; ++t) {
        const size_t idx = base + (size_t)t * HH;
        a += logf_buf[idx];
        const float hv = expf(logaddexpf_(a + carry, logv_buf[idx]));
        logf_buf[idx] = hv;   // buffer becomes h
    }
}

// ---------------------------------------------------------------------------
// K5: out = h @ Wo + bo.  Same LDS-staged tiling with a single weight matrix.
// ---------------------------------------------------------------------------
__global__ __launch_bounds__(256) void out_gemm_kernel(
    const float* __restrict__ Hbuf, const float* __restrict__ Wo,
    const float* __restrict__ bo, float* __restrict__ out)
{
    __shared__ float As[2][64][APAD];
    __shared__ float Bs[2][KC][64];

    const int lane = threadIdx.x & 31;
    const int wv   = threadIdx.x >> 5;
    const int wm   = wv >> 2;
    const int wn   = wv & 3;
    const int lm   = lane & 15;
    const int khalf = (lane >> 4) << 1;

    const int mblock = blockIdx.x * 64;
    const int col0   = blockIdx.y * 64;
    const int cn     = wn * 16 + lm;
    const int col    = col0 + cn;

    const int ar = threadIdx.x >> 2;
    const int ak = (threadIdx.x & 3) << 2;
    const int bk = threadIdx.x >> 4;
    const int bc = (threadIdx.x & 15) << 2;

    v8f acc0 = {}, acc1 = {};

    {
        *(v4f*)&As[0][ar][ak] = *(const v4f*)(Hbuf + (size_t)(mblock + ar) * HH + ak);
        *(v4f*)&Bs[0][bk][bc] = *(const v4f*)(Wo + (size_t)bk * EE + col0 + bc);
    }
    __syncthreads();

    for (int kc = 0; kc < NKC; ++kc) {
        const int cur = kc & 1;
        if (kc + 1 < NKC) {
            const int k0 = (kc + 1) * KC;
            __builtin_prefetch(Hbuf + (size_t)(mblock + ar) * HH + k0 + KC, 0, 0);
            *(v4f*)&As[cur ^ 1][ar][ak] =
                *(const v4f*)(Hbuf + (size_t)(mblock + ar) * HH + k0 + ak);
            *(v4f*)&Bs[cur ^ 1][bk][bc] =
                *(const v4f*)(Wo + (size_t)(k0 + bk) * EE + col0 + bc);
        }
#pragma unroll
        for (int kk = 0; kk < KC; kk += 4) {
            const int kh = kk + khalf;
            v2f a0 = *(const v2f*)&As[cur][wm * 32 + lm][kh];
            v2f a1 = *(const v2f*)&As[cur][wm * 32 + 16 + lm][kh];
            v2f b0;
            b0.x = Bs[cur][kh][cn];   b0.y = Bs[cur][kh + 1][cn];
            acc0 = wmma_f32(a0, b0, acc0);
            acc1 = wmma_f32(a1, b0, acc1);
        }
        __syncthreads();
    }

    const float bov = bo[col];
#pragma unroll
    for (int sub = 0; sub < 2; ++sub) {
        const v8f acc = sub ? acc1 : acc0;
        const int mbase = mblock + wm * 32 + sub * 16 + ((lane >> 4) << 3);
#pragma unroll
        for (int r = 0; r < 8; ++r) {
            out[(size_t)(mbase + r) * EE + col] = acc[r] + bov;
        }
    }
}

// ---------------------------------------------------------------------------
extern "C" void kernel_launch(void* const* d_in, const int* in_sizes, int n_in,
                              void* d_out, int out_size, void* d_ws, size_t ws_size,
                              hipStream_t stream) {
    const float* x  = (const float*)d_in[0];
    const float* h0 = (const float*)d_in[1];
    const float* Wf = (const float*)d_in[2];
    const float* bf = (const float*)d_in[3];
    const float* Wi = (const float*)d_in[4];
    const float* bi = (const float*)d_in[5];
    const float* Wh = (const float*)d_in[6];
    const float* bh = (const float*)d_in[7];
    const float* Wo = (const float*)d_in[8];
    const float* bo = (const float*)d_in[9];

    float* out   = (float*)d_out;                 // (B,S,E)
    float* hlast = out + (size_t)MM * EE;         // (B,1,H)

    float* ws       = (float*)d_ws;
    float* logf_buf = ws;                          // (B,S,H)  -> later h
    float* logv_buf = ws + BSH;                    // (B,S,H)  -> local scan
    float* sumA     = logv_buf + BSH;              // (B,NC,H)
    float* sumL     = sumA + BNH;                  // (B,NC,H)
    float* carryin  = sumL + BNH;                  // (B,NC,H)

    // K1: gate GEMMs + fused log-space epilogue (LDS-staged WMMA)
    gates_gemm_kernel<<<dim3(MM / 64, HH / 64), 256, 0, stream>>>(
        x, Wf, bf, Wi, bi, Wh, bh, logf_buf, logv_buf);

    // K2: chunk-local scans (2048 waves)
    scan_local_kernel<<<(int)(BNH / 256), 256, 0, stream>>>(
        logf_buf, logv_buf, sumA, sumL);

    // K3: scan chunk summaries; also emits h_last
    scan_chunks_kernel<<<(BB * HH) / 256, 256, 0, stream>>>(
        h0, sumA, sumL, carryin, hlast);

    // K4: apply carries -> h (in place into logf_buf)
    scan_apply_kernel<<<(int)(BNH / 256), 256, 0, stream>>>(
        logf_buf, logv_buf, carryin);

    // K5: output GEMM (LDS-staged WMMA)
    out_gemm_kernel<<<dim3(MM / 64, EE / 64), 256, 0, stream>>>(
        logf_buf, Wo, bo, out);
}